// GatherRouter_36679020708158
// MI455X (gfx1250) — compile-verified
//
#include <hip/hip_runtime.h>
#include <hip/hip_bf16.h>
#include <stdint.h>

// Problem constants (fixed by the reference): E=8, N=8192, D=1024, LOAD=16384
#define EN     65536            // E*N rows
#define DIMS   1024             // floats per row (4 KB)
#define NSLOT  16384            // output slots
#define MAXK   32               // per-slot list capacity (avg occupancy = 4)
#define NBUF   4                // LDS ring depth (issue depth 2 + safety)

// ---------------------------------------------------------------------------
// CDNA5 async global->LDS DMA (gfx1250): GLOBAL_LOAD_ASYNC_TO_LDS_B128 is
// tracked by ASYNCcnt and retires IN ORDER, drained with S_WAIT_ASYNCCNT N.
// The VDST operand is a VGPR holding the wave-relative LDS byte offset; a flat
// pointer to __shared__ has exactly that offset in its low 32 bits (ISA §10.2
// aperture mapping), so truncation yields the correct LDS address.
// ---------------------------------------------------------------------------
static __device__ __forceinline__ unsigned lds_off(const void* p) {
    return (unsigned)(unsigned long long)p;
}
static __device__ __forceinline__ void async_load_b128(unsigned ldsoff, const void* gptr) {
    asm volatile("global_load_async_to_lds_b128 %0, %1, off"
                 :: "v"(ldsoff), "v"(gptr) : "memory");
}
template <int N>
static __device__ __forceinline__ void wait_asynccnt() {
    asm volatile("s_wait_asynccnt %0" :: "i"(N) : "memory");
}

// ---------------------------------------------------------------------------
// Pass 0: zero the per-slot counters in scratch
// ---------------------------------------------------------------------------
__global__ __launch_bounds__(256) void zero_counts_kernel(int* __restrict__ counts) {
    int i = blockIdx.x * 256 + threadIdx.x;
    if (i < NSLOT) counts[i] = 0;
}

// ---------------------------------------------------------------------------
// Pass 1: invert tag -> row mapping (integer atomics only; counts are
// deterministic, list order is made deterministic by the sort in pass 2).
// ---------------------------------------------------------------------------
__global__ __launch_bounds__(256) void bin_kernel(const unsigned long long* __restrict__ tags,
                                                  int* __restrict__ counts,
                                                  int* __restrict__ lists) {
    int r = blockIdx.x * 256 + threadIdx.x;
    if (r < EN) {
        int t = (int)tags[r];                 // int64 tag, value < 16384
        int pos = atomicAdd(&counts[t], 1);
        if (pos < MAXK) lists[t * MAXK + pos] = r;
    }
}

// ---------------------------------------------------------------------------
// Pass 2: one block per output slot (256 threads = 8 wave32). Rows feeding
// this slot are DMA'd into a 4-deep LDS ring with issue depth 2, overlapping
// HBM latency with accumulation at ONE barrier per row:
//   iter k: issue row k+2 -> buf[(k+2)%4]; s_wait_asynccnt <= min(K-1-k,2);
//           __syncthreads(); accumulate buf[k%4].
// Safety: buf[(k+2)%4] was last read at iter k-2; every wave executes
// read(k-2) -> barrier(k-1) -> issue(k), so no overwrite race. In-order
// ASYNCcnt retirement makes the wait threshold sufficient for row k.
// ---------------------------------------------------------------------------
__global__ __launch_bounds__(256) void gather_kernel(const float* __restrict__ data,
                                                     const int* __restrict__ counts,
                                                     const int* __restrict__ lists,
                                                     float* __restrict__ out) {
    __shared__ int   s_cnt;
    __shared__ int   s_rows[MAXK];
    __shared__ int   s_sorted[MAXK];
    __shared__ __align__(16) float s_buf[NBUF][DIMS];   // 16 KB ring

    const int slot = blockIdx.x;
    const int tid  = threadIdx.x;

    if (tid == 0) {
        int c = counts[slot];
        s_cnt = (c < MAXK) ? c : MAXK;
    }
    if (tid < MAXK) s_rows[tid] = lists[slot * MAXK + tid];
    __syncthreads();

    const int K = s_cnt;

    // Deterministic ordering: rank-sort the (unique) row ids ascending.
    if (tid < K) {
        int r = s_rows[tid];
        int rank = 0;
        for (int j = 0; j < K; ++j) rank += (s_rows[j] < r) ? 1 : 0;
        s_sorted[rank] = r;
    }
    __syncthreads();

    const unsigned base = lds_off(&s_buf[0][0]) + (unsigned)(tid * 16);

    // Issue list-position k's row into ring slot k%4 (256 lanes x 16B = 4KB).
    auto issue = [&](int k) {
        int row = s_sorted[k];
        const char* src = (const char*)(data + (size_t)row * DIMS) + (size_t)tid * 16;
        async_load_b128(base + (unsigned)((k & (NBUF - 1)) << 12), src);
    };

    if (K > 0) issue(0);
    if (K > 1) issue(1);

    float4 acc = make_float4(0.f, 0.f, 0.f, 0.f);

    for (int k = 0; k < K; ++k) {
        if (k + 2 < K) issue(k + 2);

        const int pending = K - 1 - k;            // ops issued after row k's op
        if (pending >= 2)      wait_asynccnt<2>();
        else if (pending == 1) wait_asynccnt<1>();
        else                   wait_asynccnt<0>();
        __syncthreads();                          // all waves' slices visible

        float4 v = *reinterpret_cast<const float4*>(&s_buf[k & (NBUF - 1)][tid * 4]);
        acc.x += v.x; acc.y += v.y; acc.z += v.z; acc.w += v.w;
    }

    *reinterpret_cast<float4*>(out + (size_t)slot * DIMS + (size_t)tid * 4) = acc;
}

// ---------------------------------------------------------------------------
// Fallback path (only if d_ws is too small): zero + global f32 atomic scatter.
// ---------------------------------------------------------------------------
__global__ __launch_bounds__(256) void zero_out_kernel(float* __restrict__ out, int n) {
    int i = blockIdx.x * 256 + threadIdx.x;
    if (i < n) out[i] = 0.f;
}

__global__ __launch_bounds__(256) void scatter_atomic_kernel(const float* __restrict__ data,
                                                             const unsigned long long* __restrict__ tags,
                                                             float* __restrict__ out) {
    const int row = blockIdx.x;
    const int tid = threadIdx.x;
    const int t   = (int)tags[row];
    float4 v = *reinterpret_cast<const float4*>(data + (size_t)row * DIMS + (size_t)tid * 4);
    float* o = out + (size_t)t * DIMS + (size_t)tid * 4;
    atomicAdd(o + 0, v.x);
    atomicAdd(o + 1, v.y);
    atomicAdd(o + 2, v.z);
    atomicAdd(o + 3, v.w);
}

// ---------------------------------------------------------------------------
extern "C" void kernel_launch(void* const* d_in, const int* in_sizes, int n_in,
                              void* d_out, int out_size, void* d_ws, size_t ws_size,
                              hipStream_t stream) {
    const float* data              = (const float*)d_in[0];
    const unsigned long long* tags = (const unsigned long long*)d_in[1];
    float* out                     = (float*)d_out;

    const size_t need = (size_t)NSLOT * 4 + (size_t)NSLOT * MAXK * 4;  // ~2.1 MB

    if (ws_size >= need) {
        int* counts = (int*)d_ws;
        int* lists  = counts + NSLOT;

        zero_counts_kernel<<<NSLOT / 256, 256, 0, stream>>>(counts);
        bin_kernel<<<EN / 256, 256, 0, stream>>>(tags, counts, lists);
        gather_kernel<<<NSLOT, 256, 0, stream>>>(data, counts, lists, out);
    } else {
        const int n = NSLOT * DIMS;
        zero_out_kernel<<<(n + 255) / 256, 256, 0, stream>>>(out, n);
        scatter_atomic_kernel<<<EN, 256, 0, stream>>>(data, tags, out);
    }
}